// _DCN_Module_75892072120841
// MI455X (gfx1250) — compile-verified
//
#include <hip/hip_runtime.h>
#include <hip/hip_bf16.h>

typedef __attribute__((ext_vector_type(2))) float v2f;
typedef __attribute__((ext_vector_type(8))) float v8f;

#define DDIM 256
#define KCL  4096
#define NPTS 65536

#define NROWS   64          // embedded rows per block
#define LDSTR   258         // padded LDS row stride (floats) -> bank offset 2/row
#define MT      4           // 16-row M-tiles per wave (4*16 = 64 rows)
#define WAVES   8
#define COLS_IT 128         // 8 waves * 16 cols
#define KITERS  (KCL / COLS_IT)

// ---------------------------------------------------------------------------
// Kernel A: c2[k] = ||centers[k]||^2 ; zero the d_out accumulation regions.
// grid = K blocks x 256 threads (one block per center).
// ---------------------------------------------------------------------------
__global__ __launch_bounds__(256) void prep_kernel(
    const float* __restrict__ C, float* __restrict__ c2,
    float* __restrict__ seg_sum, int* __restrict__ seg_cnt,
    float* __restrict__ loss_acc) {
  const int k = blockIdx.x;
  const int d = threadIdx.x;
  const size_t idx = (size_t)k * DDIM + d;

  seg_sum[idx] = 0.0f;                 // centers-output region: accumulator
  if (d == 0) seg_cnt[k] = 0;          // counts-output region: accumulator
  if (k == 0 && d == 0) *loss_acc = 0.0f;

  __shared__ float red[256];
  const float v = C[idx];
  red[d] = v * v;
  __syncthreads();
  #pragma unroll
  for (int s = 128; s > 0; s >>= 1) {
    if (d < s) red[d] += red[d + s];
    __syncthreads();
  }
  if (d == 0) c2[k] = red[0];
}

// ---------------------------------------------------------------------------
// Kernel B: fused distance-GEMM (fp32 WMMA 16x16x4) + argmin + loss +
// segment-sum atomics. grid = N/64 blocks x 256 threads (8 wave32s).
// ---------------------------------------------------------------------------
__global__ __launch_bounds__(256) void assign_kernel(
    const float* __restrict__ E, const float* __restrict__ C,
    const float* __restrict__ c2, int* __restrict__ labels_out,
    float* __restrict__ seg_sum, int* __restrict__ seg_cnt,
    float* __restrict__ loss_acc) {
  __shared__ float As[NROWS * LDSTR];
  __shared__ float x2s[NROWS];
  __shared__ float wBestV[WAVES][NROWS];
  __shared__ int   wBestI[WAVES][NROWS];
  __shared__ int   labs[NROWS];

  const int tid  = threadIdx.x;
  const int wave = tid >> 5;
  const int lane = tid & 31;
  const int hi   = lane >> 4;     // half-wave select (k-pair 0..1 vs 2..3)
  const int ln   = lane & 15;     // M (for A) / N (for B) within tile
  const int rowBase = blockIdx.x * NROWS;

  // Stage the 64x256 A-tile into LDS (padded rows), coalesced float4 loads.
  for (int i = tid; i < NROWS * (DDIM / 4); i += 256) {
    const int r  = i / (DDIM / 4);
    const int c4 = i % (DDIM / 4);
    const float4 v = ((const float4*)(E + (size_t)(rowBase + r) * DDIM))[c4];
    float* dst = &As[r * LDSTR + c4 * 4];
    dst[0] = v.x; dst[1] = v.y; dst[2] = v.z; dst[3] = v.w;
  }
  __syncthreads();

  // Per-row ||x||^2 (needed only for the loss term).
  if (tid < NROWS) {
    const float* row = &As[tid * LDSTR];
    float s = 0.0f;
    #pragma unroll 8
    for (int d = 0; d < DDIM; ++d) s += row[d] * row[d];
    x2s[tid] = s;
  }

  float bestV[MT][8];
  int   bestI[MT][8];
  #pragma unroll
  for (int t = 0; t < MT; ++t)
    #pragma unroll
    for (int r = 0; r < 8; ++r) { bestV[t][r] = 3.4e38f; bestI[t][r] = 0; }

  for (int it = 0; it < KITERS; ++it) {
    const int myCol = it * COLS_IT + wave * 16 + ln;
    const float c2col = c2[myCol];
    const float* Bp = C + (size_t)myCol * DDIM + 2 * hi;

    v8f acc[MT];
    #pragma unroll
    for (int t = 0; t < MT; ++t) acc[t] = (v8f){};

    #pragma unroll 4
    for (int d = 0; d < DDIM; d += 4) {
      // B (4x16) fragment: lane holds B[2hi..2hi+1][ln] = centers[col][d+2hi..]
      const v2f b = *(const v2f*)(Bp + d);
      #pragma unroll
      for (int t = 0; t < MT; ++t) {
        // A (16x4) fragment: lane holds A[ln][d+2hi..d+2hi+1]
        const v2f a = *(const v2f*)(&As[(t * 16 + ln) * LDSTR + d + 2 * hi]);
        acc[t] = __builtin_amdgcn_wmma_f32_16x16x4_f32(
            false, a, false, b, (short)0, acc[t], false, false);
      }
    }

    // score = c2[col] - 2*dot ; running per-lane argmin (x2 row-constant).
    #pragma unroll
    for (int t = 0; t < MT; ++t)
      #pragma unroll
      for (int r = 0; r < 8; ++r) {
        const float v = c2col - 2.0f * acc[t][r];
        if (v < bestV[t][r]) { bestV[t][r] = v; bestI[t][r] = myCol; }
      }
  }

  // Reduce argmin across the 16 lanes of each half-wave (N dimension).
  #pragma unroll
  for (int t = 0; t < MT; ++t)
    #pragma unroll
    for (int r = 0; r < 8; ++r) {
      float v = bestV[t][r];
      int   i = bestI[t][r];
      #pragma unroll
      for (int m = 8; m >= 1; m >>= 1) {
        const float ov = __shfl_xor(v, m, 32);
        const int   oi = __shfl_xor(i, m, 32);
        if (ov < v || (ov == v && oi < i)) { v = ov; i = oi; }
      }
      if (ln == 0) {
        const int row = t * 16 + hi * 8 + r;   // C/D layout: M = vgpr + 8*hi
        wBestV[wave][row] = v;
        wBestI[wave][row] = i;
      }
    }
  __syncthreads();

  // Reduce across the 8 waves; emit label, loss, count.
  if (tid < NROWS) {
    float bv = wBestV[0][tid];
    int   bi = wBestI[0][tid];
    #pragma unroll
    for (int w = 1; w < WAVES; ++w) {
      const float v = wBestV[w][tid];
      const int   i = wBestI[w][tid];
      if (v < bv || (v == bv && i < bi)) { bv = v; bi = i; }
    }
    labs[tid] = bi;
    labels_out[rowBase + tid] = bi;
    atomicAdd(loss_acc, x2s[tid] + bv);   // d2 = x2 + (c2 - 2 dot)
    atomicAdd(&seg_cnt[bi], 1);
  }
  __syncthreads();

  // Segment-sum straight from the LDS A-tile (embedded read exactly once).
  for (int r = 0; r < NROWS; ++r) {
    atomicAdd(&seg_sum[(size_t)labs[r] * DDIM + tid], As[r * LDSTR + tid]);
  }
}

// ---------------------------------------------------------------------------
// Kernel C: in-place closed-form centroid update + counts + loss/N.
// grid = K blocks x 256 threads.
// ---------------------------------------------------------------------------
__global__ __launch_bounds__(256) void finalize_kernel(
    const float* __restrict__ C, const int* __restrict__ counts_in,
    float* __restrict__ centers_io, int* __restrict__ counts_io,
    float* __restrict__ loss_io) {
  const int k = blockIdx.x;
  const int d = threadIdx.x;
  const int m = counts_io[k];                    // seg_cnt (accumulated)
  const float c0 = (float)counts_in[k];
  const size_t idx = (size_t)k * DDIM + d;
  centers_io[idx] = (c0 * C[idx] + centers_io[idx]) / (c0 + (float)m);
  __syncthreads();
  if (d == 0) {
    counts_io[k] = counts_in[k] + m;
    if (k == 0) loss_io[0] = loss_io[0] / (float)NPTS;
  }
}

// ---------------------------------------------------------------------------
extern "C" void kernel_launch(void* const* d_in, const int* in_sizes, int n_in,
                              void* d_out, int out_size, void* d_ws, size_t ws_size,
                              hipStream_t stream) {
  const float* E      = (const float*)d_in[0];   // [N, D]
  const float* C      = (const float*)d_in[1];   // [K, D]
  const int*   counts = (const int*)d_in[2];     // [K]

  // d_out layout (flat, return order): labels[N] i32 | centers[K*D] f32 |
  // counts[K] i32 | loss[1] f32
  int*   out_labels  = (int*)d_out;
  float* out_centers = (float*)d_out + NPTS;
  int*   out_counts  = (int*)((float*)d_out + NPTS + (size_t)KCL * DDIM);
  float* out_loss    = (float*)d_out + NPTS + (size_t)KCL * DDIM + KCL;

  float* c2 = (float*)d_ws;                      // K floats of scratch

  prep_kernel<<<KCL, 256, 0, stream>>>(C, c2, out_centers, out_counts, out_loss);
  assign_kernel<<<NPTS / NROWS, 256, 0, stream>>>(E, C, c2, out_labels,
                                                  out_centers, out_counts, out_loss);
  finalize_kernel<<<KCL, 256, 0, stream>>>(C, counts, out_centers, out_counts, out_loss);
}